// ClassicalVMD_44461501448550
// MI455X (gfx1250) — compile-verified
//
#include <hip/hip_runtime.h>
#include <math.h>

// ============================================================================
// VMD for MI455X / gfx1250 (wave32, WMMA, async global<->LDS).
//
//  Kernel 1: forward 4096-pt FFT per signal (128 blocks x 512 thr), radix-16^3
//            via V_WMMA_F32_16X16X4_F32 complex tile matmuls in LDS.
//            Input gather + spectrum scatter via GLOBAL_*_ASYNC_*_LDS_B32.
//  Kernel 2: 100 VMD iterations, one block per signal, u-state in registers,
//            9-scalar block reduction per iteration, uniform early exit.
//  Kernel 3: inverse FFT per (signal,mode) (512 blocks), Hermitian extension
//            with 1/N scale folded in, async LDS->global scatter of the result.
// ============================================================================

typedef __attribute__((ext_vector_type(2))) float v2f;
typedef __attribute__((ext_vector_type(8))) float v8f;

#define N_SIG 128
#define T_LEN 4096
#define FR    2049
#define KM    4
#define TWO_PI 6.283185307179586f

// ---- CDNA5 async global<->LDS helpers (ASYNCcnt-tracked DMA, no VGPR data) --
__device__ __forceinline__ void async_g2l_b32(void* lds_dst, const void* gsrc) {
    unsigned loff = (unsigned)(size_t)lds_dst;              // LDS byte offset
    unsigned long long ga = (unsigned long long)(size_t)gsrc;
    asm volatile("global_load_async_to_lds_b32 %0, %1, off"
                 :: "v"(loff), "v"(ga) : "memory");
}
__device__ __forceinline__ void async_l2g_b32(void* gdst, const void* lds_src) {
    unsigned loff = (unsigned)(size_t)lds_src;
    unsigned long long ga = (unsigned long long)(size_t)gdst;
    asm volatile("global_store_async_from_lds_b32 %0, %1, off"
                 :: "v"(ga), "v"(loff) : "memory");
}
__device__ __forceinline__ void wait_async0() {
    asm volatile("s_wait_asynccnt 0" ::: "memory");
}

// ----------------------------------------------------------------------------
// One radix-16 stage: for this wave's tile, compute O = F16 x D (complex,
// 16x16), apply twiddles, write back in place.
//   B element (k,n) read at  lre/lim[kStride*k + nStride*n + base]
//   D element (M,n) written  lre/lim[oRow*M + oCol*n   + base]
// TW: 0 -> W256^{M*tile}, 1 -> W4096^{(tile+16M)*n}, 2 -> none.
// WMMA fp32 16x16x4 lane layout (ISA 7.12.2): lanes 0-15 hold K={0,1} in the
// two VGPRs, lanes 16-31 hold K={2,3}; C/D vgpr i holds row M=i (+8 for the
// upper half-wave), column N = lane&15.
// ----------------------------------------------------------------------------
template <int SIGN, int TW>
__device__ __forceinline__ void fft_stage(float* lre, float* lim,
                                          const v2f (&fr)[4], const v2f (&fi)[4],
                                          const v2f (&fin)[4],
                                          int kStride, int nStride, int base,
                                          int oRow, int oCol,
                                          int tile, int lane) {
    const int half = (lane < 16) ? 0 : 2;   // K-offset for this half-wave
    const int r    = lane & 15;             // B column / output column

    // Load ALL B chunks (re+im) into VGPRs before any store (in-place safety:
    // this wave writes exactly the element set it reads).
    v2f br[4], bi[4];
#pragma unroll
    for (int kk = 0; kk < 4; ++kk) {
        int k0 = 4 * kk + half;
        int i0 = kStride * k0 + nStride * r + base;
        int i1 = i0 + kStride;
        br[kk].x = lre[i0]; br[kk].y = lre[i1];
        bi[kk].x = lim[i0]; bi[kk].y = lim[i1];
    }

    // Complex matmul: Re = Fr*Br - Fi*Bi ; Im = Fr*Bi + Fi*Br  (16 WMMAs).
    v8f accr = {0.f,0.f,0.f,0.f,0.f,0.f,0.f,0.f};
    v8f acci = {0.f,0.f,0.f,0.f,0.f,0.f,0.f,0.f};
#pragma unroll
    for (int kk = 0; kk < 4; ++kk)
        accr = __builtin_amdgcn_wmma_f32_16x16x4_f32(false, fr[kk],  false, br[kk],
                                                     (short)0, accr, false, false);
#pragma unroll
    for (int kk = 0; kk < 4; ++kk)
        accr = __builtin_amdgcn_wmma_f32_16x16x4_f32(false, fin[kk], false, bi[kk],
                                                     (short)0, accr, false, false);
#pragma unroll
    for (int kk = 0; kk < 4; ++kk)
        acci = __builtin_amdgcn_wmma_f32_16x16x4_f32(false, fr[kk],  false, bi[kk],
                                                     (short)0, acci, false, false);
#pragma unroll
    for (int kk = 0; kk < 4; ++kk)
        acci = __builtin_amdgcn_wmma_f32_16x16x4_f32(false, fi[kk],  false, br[kk],
                                                     (short)0, acci, false, false);

    // Twiddle (incremental rotation over M) + writeback.
    const int mOff = (lane < 16) ? 0 : 8;
    float twr = 1.f, twi = 0.f, str = 1.f, sti = 0.f;
    if (TW == 0) {
        float th = (float)SIGN * (TWO_PI / 256.f) * (float)tile;
        str = cosf(th); sti = sinf(th);
        float a0 = th * (float)mOff;
        twr = cosf(a0); twi = sinf(a0);
    } else if (TW == 1) {
        float th = (float)SIGN * (TWO_PI / 256.f) * (float)r;   // step: 2*pi*16*n/4096
        str = cosf(th); sti = sinf(th);
        float a0 = (float)SIGN * (TWO_PI / 4096.f) * (float)(r * (tile + 16 * mOff));
        twr = cosf(a0); twi = sinf(a0);
    }
#pragma unroll
    for (int i = 0; i < 8; ++i) {
        int M = i + mOff;
        float orr = accr[i], oii = acci[i];
        if (TW != 2) {
            float tr = orr * twr - oii * twi;
            oii = orr * twi + oii * twr;
            orr = tr;
            float nr = twr * str - twi * sti;  // advance rotation: tw *= step
            twi = twr * sti + twi * str;
            twr = nr;
        }
        int s = oRow * M + oCol * r + base;
        lre[s] = orr;
        lim[s] = oii;
    }
}

// Full 4096-pt complex FFT in LDS (natural-order in, base-16 digit-reversed
// out: X[j0+16j1+256j2] lands at index j0*256+j1*16+j2). 512 threads.
template <int SIGN>
__device__ __forceinline__ void fft4096(float* lre, float* lim, int tid) {
    const int wave = tid >> 5;
    const int lane = tid & 31;
    const int half = (lane < 16) ? 0 : 2;
    const int r    = lane & 15;

    // DFT-16 matrix chunks (stage-invariant): F[r][c] = exp(SIGN*2*pi*i*r*c/16)
    v2f fr[4], fi[4], fin[4];
    const float w16 = TWO_PI / 16.0f;
#pragma unroll
    for (int kk = 0; kk < 4; ++kk) {
        int c0 = 4 * kk + half;
        float a0 = w16 * (float)((r * c0) & 15);
        float a1 = w16 * (float)((r * (c0 + 1)) & 15);
        fr[kk].x = cosf(a0);               fr[kk].y = cosf(a1);
        fi[kk].x = (float)SIGN * sinf(a0); fi[kk].y = (float)SIGN * sinf(a1);
        fin[kk].x = -fi[kk].x;             fin[kk].y = -fi[kk].y;
    }

    __syncthreads();
    fft_stage<SIGN, 0>(lre, lim, fr, fi, fin, 256, 1, wave * 16,  256, 1, wave, lane);
    __syncthreads();
    fft_stage<SIGN, 1>(lre, lim, fr, fi, fin, 16,  1, wave * 256, 16,  1, wave, lane);
    __syncthreads();
    fft_stage<SIGN, 2>(lre, lim, fr, fi, fin, 1,  16, wave * 256, 1,  16, wave, lane);
    __syncthreads();
}

__device__ __forceinline__ int digit_rev(int j) {
    return ((j & 15) << 8) | (j & 0xF0) | (j >> 8);
}

// ----------------------------------------------------------------------------
// Kernel 1: async strided gather of signal n=(b,c), forward FFT, async store
// of the 2049 rfft bins (digit-reversal folded into the LDS-side address).
// ----------------------------------------------------------------------------
__global__ __launch_bounds__(512) void vmd_fft_fwd(const float* __restrict__ x,
                                                   float* __restrict__ f_re,
                                                   float* __restrict__ f_im) {
    __shared__ float lre[T_LEN];
    __shared__ float lim[T_LEN];
    const int n = blockIdx.x, tid = threadIdx.x;
    const int b = n >> 3, c = n & 7;
    const float* src = x + ((size_t)b * T_LEN * 8 + c);
#pragma unroll
    for (int i = 0; i < 8; ++i) {
        int t = tid + i * 512;
        async_g2l_b32(&lre[t], src + (size_t)t * 8);   // DMA, ASYNCcnt
        lim[t] = 0.f;
    }
    wait_async0();
    fft4096<-1>(lre, lim, tid);
    for (int j = tid; j < FR; j += 512) {
        int s = digit_rev(j);
        async_l2g_b32(f_re + n * FR + j, &lre[s]);
        async_l2g_b32(f_im + n * FR + j, &lim[s]);
    }
    wait_async0();
}

// ----------------------------------------------------------------------------
// Kernel 2: 100 VMD iterations, block per signal, state in registers.
// ----------------------------------------------------------------------------
__global__ __launch_bounds__(512) void vmd_iterate(const float* __restrict__ f_re,
                                                   const float* __restrict__ f_im,
                                                   float* __restrict__ u_re,
                                                   float* __restrict__ u_im) {
    const int n = blockIdx.x, tid = threadIdx.x;
    const int lane = tid & 31, wave = tid >> 5;
    __shared__ float sPart[16][9];
    __shared__ float sOm[KM];
    __shared__ int sCont;

    float fhr[5], fhi[5], fq[5];
    float ur[5][KM], ui[5][KM];
#pragma unroll
    for (int i = 0; i < 5; ++i) {
        int bin = tid + i * 512;
        bool ok = bin < FR;
        fhr[i] = ok ? f_re[n * FR + bin] : 0.f;
        fhi[i] = ok ? f_im[n * FR + bin] : 0.f;
        fq[i]  = (float)bin * (0.5f / 2048.f);
#pragma unroll
        for (int k = 0; k < KM; ++k) { ur[i][k] = 0.f; ui[i][k] = 0.f; }
    }
    float om[KM] = {0.05f, 0.05f + 0.4f / 3.f, 0.05f + 0.8f / 3.f, 0.45f};

    for (int it = 0; it < 100; ++it) {
        float psum[KM] = {0.f, 0.f, 0.f, 0.f};
        float wsum[KM] = {0.f, 0.f, 0.f, 0.f};
        float dsum = 0.f;
#pragma unroll
        for (int i = 0; i < 5; ++i) {
            int bin = tid + i * 512;
            if (bin >= FR) continue;
            float sr = 0.f, si = 0.f;
#pragma unroll
            for (int k = 0; k < KM; ++k) { sr += ur[i][k]; si += ui[i][k]; }
#pragma unroll
            for (int k = 0; k < KM; ++k) {
                float nr = fhr[i] - (sr - ur[i][k]);      // lambda == 0 (TAU=0)
                float ni = fhi[i] - (si - ui[i][k]);
                float d  = fq[i] - om[k];
                float inv = 1.f / (1.f + 4000.f * d * d); // 2*ALPHA = 4000
                nr *= inv; ni *= inv;
                float pw = nr * nr + ni * ni;
                psum[k] += pw;
                wsum[k] += pw * fq[i];
                float dr = nr - ur[i][k], di = ni - ui[i][k];
                dsum += dr * dr + di * di;
                ur[i][k] = nr; ui[i][k] = ni;
            }
        }
        // wave32 butterfly reduction of the 9 partials
#pragma unroll
        for (int off = 16; off > 0; off >>= 1) {
#pragma unroll
            for (int k = 0; k < KM; ++k) {
                psum[k] += __shfl_xor(psum[k], off, 32);
                wsum[k] += __shfl_xor(wsum[k], off, 32);
            }
            dsum += __shfl_xor(dsum, off, 32);
        }
        if (lane == 0) {
#pragma unroll
            for (int k = 0; k < KM; ++k) {
                sPart[wave][k]     = psum[k];
                sPart[wave][4 + k] = wsum[k];
            }
            sPart[wave][8] = dsum;
        }
        __syncthreads();
        if (tid == 0) {
            float P[KM] = {0.f, 0.f, 0.f, 0.f}, W[KM] = {0.f, 0.f, 0.f, 0.f}, D = 0.f;
            for (int w = 0; w < 16; ++w) {
#pragma unroll
                for (int k = 0; k < KM; ++k) { P[k] += sPart[w][k]; W[k] += sPart[w][4 + k]; }
                D += sPart[w][8];
            }
#pragma unroll
            for (int k = 0; k < KM; ++k)
                sOm[k] = (P[k] > 1e-12f) ? (W[k] / fmaxf(P[k], 1e-30f)) : om[k];
            sCont = (D >= 1e-7f) ? 1 : 0;
        }
        __syncthreads();
#pragma unroll
        for (int k = 0; k < KM; ++k) om[k] = sOm[k];
        int cont = sCont;
        __syncthreads();
        if (!cont) break;   // uniform per block == per-signal act mask
    }
#pragma unroll
    for (int i = 0; i < 5; ++i) {
        int bin = tid + i * 512;
        if (bin >= FR) continue;
#pragma unroll
        for (int k = 0; k < KM; ++k) {
            u_re[((size_t)(n * KM + k)) * FR + bin] = ur[i][k];
            u_im[((size_t)(n * KM + k)) * FR + bin] = ui[i][k];
        }
    }
}

// ----------------------------------------------------------------------------
// Kernel 3: Hermitian extension (1/N scale folded in) + inverse FFT per
// (signal,mode); async LDS->global scatter of the real part.
// ----------------------------------------------------------------------------
__global__ __launch_bounds__(512) void vmd_ifft(const float* __restrict__ u_re,
                                                const float* __restrict__ u_im,
                                                float* __restrict__ out) {
    __shared__ float lre[T_LEN];
    __shared__ float lim[T_LEN];
    const int bk = blockIdx.x;            // n*4 + k
    const int n = bk >> 2, k = bk & 3;
    const int tid = threadIdx.x;
    const float* ure = u_re + (size_t)bk * FR;
    const float* uim = u_im + (size_t)bk * FR;
    const float sc = 1.f / 4096.f;        // irfft scale, folded in by linearity
#pragma unroll
    for (int i = 0; i < 8; ++i) {
        int j = tid + i * 512;
        if (j <= 2048) { lre[j] = ure[j] * sc; lim[j] = uim[j] * sc; }
        else { int jj = T_LEN - j; lre[j] = ure[jj] * sc; lim[j] = -uim[jj] * sc; }
    }
    fft4096<1>(lre, lim, tid);
    const int b = n >> 3, c = n & 7;
    float* dst = out + ((size_t)b * KM * T_LEN * 8 + (size_t)k * T_LEN * 8 + c);
#pragma unroll
    for (int i = 0; i < 8; ++i) {
        int t = tid + i * 512;
        int s = digit_rev(t);
        async_l2g_b32(dst + (size_t)t * 8, &lre[s]);
    }
    wait_async0();
}

// ----------------------------------------------------------------------------
extern "C" void kernel_launch(void* const* d_in, const int* in_sizes, int n_in,
                              void* d_out, int out_size, void* d_ws, size_t ws_size,
                              hipStream_t stream) {
    (void)in_sizes; (void)n_in; (void)out_size; (void)ws_size;
    const float* x = (const float*)d_in[0];
    float* out = (float*)d_out;

    // Workspace layout (floats): f_re|f_im: 128*2049 each; u_re|u_im: 128*4*2049 each
    float* ws   = (float*)d_ws;
    float* f_re = ws;
    float* f_im = f_re + (size_t)N_SIG * FR;
    float* u_re = f_im + (size_t)N_SIG * FR;
    float* u_im = u_re + (size_t)N_SIG * KM * FR;

    vmd_fft_fwd<<<N_SIG, 512, 0, stream>>>(x, f_re, f_im);
    vmd_iterate<<<N_SIG, 512, 0, stream>>>(f_re, f_im, u_re, u_im);
    vmd_ifft<<<N_SIG * KM, 512, 0, stream>>>(u_re, u_im, out);
}